// qVectorField_65695819759871
// MI455X (gfx1250) — compile-verified
//
#include <hip/hip_runtime.h>
#include <math.h>

typedef __attribute__((ext_vector_type(2))) float v2f;
typedef __attribute__((ext_vector_type(8))) float v8f;

#define SIGMA0f 0.001f

// ---------------------------------------------------------------------------
// Kernel 1: precompute ||x0_m||^2 for all M rows into workspace.
// ---------------------------------------------------------------------------
__global__ void x0_norms_kernel(const float* __restrict__ x0,
                                float* __restrict__ nrm, int M) {
  int m = blockIdx.x * blockDim.x + threadIdx.x;
  if (m >= M) return;
  const float4* p = reinterpret_cast<const float4*>(x0 + (size_t)m * 64);
  float s = 0.f;
#pragma unroll
  for (int i = 0; i < 16; ++i) {
    float4 v = p[i];
    s += v.x * v.x + v.y * v.y + v.z * v.z + v.w * v.w;
  }
  nrm[m] = s;
}

// ---------------------------------------------------------------------------
// Kernel 2: fused flash-attention-style pass.
//   Grid: B/16 blocks, 128 threads (4 waves).  Wave w owns m-strip
//   [it*64 + w*16, +16) each iteration; online softmax per wave, merged at end.
// WMMA f32 16x16x4 layouts (wave32):
//   A 16x4 : lanes0-15 row=lane  (v0:K0, v1:K1); lanes16-31 row=lane-16 (K2,K3)
//   B  4x16: lanes0-15 col=lane  (v0:K0, v1:K1); lanes16-31 col=lane-16 (K2,K3)
//   C 16x16: VGPR j holds row j (lanes0-15) / row j+8 (lanes16-31), col=lane&15
// ---------------------------------------------------------------------------
__launch_bounds__(128)
__global__ void qflow_wmma_kernel(const float* __restrict__ t,
                                  const float* __restrict__ x,
                                  const float* __restrict__ x0,
                                  const float* __restrict__ nrm,
                                  float* __restrict__ out, int M) {
  __shared__ float p_lds[4][16 * 18];  // per-wave exp-weight tile (padded)
  __shared__ float m_lds[4][16];
  __shared__ float s_lds[4][16];
  __shared__ float acc_lds[16 * 64];
  __shared__ float stot_lds[16];

  const int lane = threadIdx.x & 31;
  const int wave = threadIdx.x >> 5;
  const int n    = lane & 15;   // column within 16-tile
  const int hi   = lane >> 4;   // half-wave select
  const int base = hi * 8;      // C-layout row base for this lane
  const int r0   = blockIdx.x * 16;

  // Per-row softmax-affine constants for this lane's 8 rows.
  float c1[8], c2[8];
#pragma unroll
  for (int j = 0; j < 8; ++j) {
    float tv = t[r0 + base + j];
    float al = 1.f - tv;
    float sg = SIGMA0f + (1.f - SIGMA0f) * tv;
    float iv = 1.f / (sg * sg);
    c1[j] = al * iv;            // scales <x, x0>
    c2[j] = 0.5f * al * al * iv; // scales ||x0||^2
  }

  // A-tile: x rows r0..r0+15 in WMMA-A layout, K=64 -> 16 chunks of v2f.
  v2f a[16];
  {
    const float* xr = x + (size_t)(r0 + n) * 64 + 2 * hi;
#pragma unroll
    for (int k = 0; k < 16; ++k)
      a[k] = *reinterpret_cast<const v2f*>(xr + 4 * k);
  }

  v8f acc0 = {}, acc1 = {}, acc2 = {}, acc3 = {};  // qt accum, 16x64 C-layout
  float mrun[8], srun[8];
#pragma unroll
  for (int j = 0; j < 8; ++j) { mrun[j] = -INFINITY; srun[j] = 0.f; }

  float* pl = &p_lds[wave][0];
  const int iters = M >> 6;  // M / 64 (4 waves * 16 columns)

  for (int it = 0; it < iters; ++it) {
    const int m0 = it * 64 + wave * 16;

    // ---- GEMM1: S = x_tile . x0_tile^T  (16x16, K=64) ----
    v8f s8 = {};
    {
      const float* br = x0 + (size_t)(m0 + n) * 64 + 2 * hi;
#pragma unroll
      for (int k = 0; k < 16; ++k) {
        v2f b = *reinterpret_cast<const v2f*>(br + 4 * k);
        s8 = __builtin_amdgcn_wmma_f32_16x16x4_f32(false, a[k], false, b,
                                                   (short)0, s8, false, false);
      }
    }

    const float nr = nrm[m0 + n];

    // ---- logits + online softmax (segmented 16-lane reductions) ----
    float pv[8], fs[8];
#pragma unroll
    for (int j = 0; j < 8; ++j) {
      float l = c1[j] * s8[j] - c2[j] * nr;
      float rm = l;
      rm = fmaxf(rm, __shfl_xor(rm, 1, 32));
      rm = fmaxf(rm, __shfl_xor(rm, 2, 32));
      rm = fmaxf(rm, __shfl_xor(rm, 4, 32));
      rm = fmaxf(rm, __shfl_xor(rm, 8, 32));
      float mn = fmaxf(mrun[j], rm);
      float sc = __expf(mrun[j] - mn);
      float pe = __expf(l - mn);
      float rs = pe;
      rs += __shfl_xor(rs, 1, 32);
      rs += __shfl_xor(rs, 2, 32);
      rs += __shfl_xor(rs, 4, 32);
      rs += __shfl_xor(rs, 8, 32);
      srun[j] = srun[j] * sc + rs;
      mrun[j] = mn;
      pv[j] = pe;
      fs[j] = sc;
    }

    // Rescale accumulator rows, stage exp-weights to LDS (C-layout -> A-layout).
#pragma unroll
    for (int j = 0; j < 8; ++j) {
      acc0[j] *= fs[j]; acc1[j] *= fs[j];
      acc2[j] *= fs[j]; acc3[j] *= fs[j];
      pl[(base + j) * 18 + n] = pv[j];
    }
    asm volatile("s_wait_dscnt 0" ::: "memory");  // own-wave LDS RAW

    v2f pa[4];
#pragma unroll
    for (int k2 = 0; k2 < 4; ++k2)
      pa[k2] = *reinterpret_cast<const v2f*>(pl + n * 18 + 4 * k2 + 2 * hi);

    // ---- GEMM2: acc += P(16x16) . x0_tile(16x64), 4 d-tiles x K=16 ----
#pragma unroll
    for (int k2 = 0; k2 < 4; ++k2) {
      const float* bp = x0 + (size_t)(m0 + 4 * k2 + 2 * hi) * 64 + n;
      v2f b;
      b.x = bp[0];   b.y = bp[64];
      acc0 = __builtin_amdgcn_wmma_f32_16x16x4_f32(false, pa[k2], false, b,
                                                   (short)0, acc0, false, false);
      b.x = bp[16];  b.y = bp[80];
      acc1 = __builtin_amdgcn_wmma_f32_16x16x4_f32(false, pa[k2], false, b,
                                                   (short)0, acc1, false, false);
      b.x = bp[32];  b.y = bp[96];
      acc2 = __builtin_amdgcn_wmma_f32_16x16x4_f32(false, pa[k2], false, b,
                                                   (short)0, acc2, false, false);
      b.x = bp[48];  b.y = bp[112];
      acc3 = __builtin_amdgcn_wmma_f32_16x16x4_f32(false, pa[k2], false, b,
                                                   (short)0, acc3, false, false);
    }
  }

  // ---- cross-wave merge (log-sum-exp combine) ----
  if (n == 0) {  // lanes 0 and 16 hold the two row halves
#pragma unroll
    for (int j = 0; j < 8; ++j) {
      m_lds[wave][base + j] = mrun[j];
      s_lds[wave][base + j] = srun[j];
    }
  }
  for (int i = threadIdx.x; i < 16 * 64; i += 128) acc_lds[i] = 0.f;
  __syncthreads();

  float fmy[8];
#pragma unroll
  for (int j = 0; j < 8; ++j) {
    int r = base + j;
    float Mx = fmaxf(fmaxf(m_lds[0][r], m_lds[1][r]),
                     fmaxf(m_lds[2][r], m_lds[3][r]));
    float st = s_lds[0][r] * __expf(m_lds[0][r] - Mx) +
               s_lds[1][r] * __expf(m_lds[1][r] - Mx) +
               s_lds[2][r] * __expf(m_lds[2][r] - Mx) +
               s_lds[3][r] * __expf(m_lds[3][r] - Mx);
    fmy[j] = __expf(mrun[j] - Mx);
    if (wave == 0 && n == 0) stot_lds[r] = st;
  }
#pragma unroll
  for (int j = 0; j < 8; ++j) {
    int r = base + j;
    float f = fmy[j];
    atomicAdd(&acc_lds[r * 64 + n],      acc0[j] * f);
    atomicAdd(&acc_lds[r * 64 + 16 + n], acc1[j] * f);
    atomicAdd(&acc_lds[r * 64 + 32 + n], acc2[j] * f);
    atomicAdd(&acc_lds[r * 64 + 48 + n], acc3[j] * f);
  }
  __syncthreads();

#pragma unroll
  for (int q = 0; q < 8; ++q) {
    int idx = threadIdx.x + 128 * q;
    int r = idx >> 6;
    out[(size_t)(r0 + r) * 64 + (idx & 63)] = acc_lds[idx] / stot_lds[r];
  }
}

// ---------------------------------------------------------------------------
extern "C" void kernel_launch(void* const* d_in, const int* in_sizes, int n_in,
                              void* d_out, int out_size, void* d_ws,
                              size_t ws_size, hipStream_t stream) {
  const float* t  = (const float*)d_in[0];   // (B,1)
  const float* x  = (const float*)d_in[1];   // (B,64)
  const float* x0 = (const float*)d_in[2];   // (M,64)
  float* out = (float*)d_out;                // (B,64)
  float* nrm = (float*)d_ws;                 // M floats

  const int B = in_sizes[0];
  const int M = in_sizes[2] / 64;

  x0_norms_kernel<<<(M + 255) / 256, 256, 0, stream>>>(x0, nrm, M);
  qflow_wmma_kernel<<<B / 16, 128, 0, stream>>>(t, x, x0, nrm, out, M);
}